// SP_70600672411774
// MI455X (gfx1250) — compile-verified
//
#include <hip/hip_runtime.h>
#include <hip/hip_bf16.h>
#include <math.h>

typedef __attribute__((ext_vector_type(4))) float v4f;
typedef __attribute__((ext_vector_type(2))) float v2f;
typedef __attribute__((ext_vector_type(8))) float v8f;

#define B_   64
#define T_   2048
#define D_   768
#define S_   16                 // T-segments
#define TSEG (T_ / S_)          // 128

// ws layout (in floats):
//   [0, S*B*D)         partial sums
//   [S*B*D, 2*S*B*D)   partial sums of squares
//   then B ints        lengths
#define PSQ_OFS  ((size_t)S_ * B_ * D_)
#define LEN_OFS  (2u * S_ * B_ * D_)

// ---------------------------------------------------------------------------
// Pass 1: per-batch length = first_negative_index + 1, else T.
// ---------------------------------------------------------------------------
__global__ void SP_lengths_kernel(const float* __restrict__ mask,
                                  int* __restrict__ lens) {
    __shared__ int smin[256];
    const int b   = blockIdx.x;
    const int tid = threadIdx.x;
    const float* m = mask + (size_t)b * T_;
    int local = T_;
    for (int t = tid; t < T_; t += 256) {
        if (m[t] < 0.0f && t < local) local = t;
    }
    smin[tid] = local;
    __syncthreads();
    for (int off = 128; off > 0; off >>= 1) {
        if (tid < off) smin[tid] = min(smin[tid], smin[tid + off]);
        __syncthreads();
    }
    if (tid == 0) {
        const int fn = smin[0];
        lens[b] = (fn == T_) ? T_ : fn + 1;   // first_neg + 1, or T
    }
}

// ---------------------------------------------------------------------------
// Pass 2: streaming partial sums. One block per (segment, batch); 192 threads
// (6 wave32), each owning 4 contiguous d's -> coalesced 128-bit NT loads.
// Mask is a prefix mask, so we just clamp the t-loop to length.
// ---------------------------------------------------------------------------
__global__ void SP_partials_kernel(const float* __restrict__ feat,
                                   const int* __restrict__ lens,
                                   float* __restrict__ psum,
                                   float* __restrict__ psq) {
    const int seg = blockIdx.x;
    const int b   = blockIdx.y;
    const int j   = threadIdx.x;          // 0..191 -> d = 4j .. 4j+3
    const int len = lens[b];
    const int t0  = seg * TSEG;
    const int t1  = min(t0 + TSEG, len);

    v4f s = {0.f, 0.f, 0.f, 0.f};
    v4f q = {0.f, 0.f, 0.f, 0.f};
    const float* base = feat + ((size_t)(b * T_ + t0)) * D_ + 4 * j;
    for (int t = t0; t < t1; ++t) {
        const v4f v = __builtin_nontemporal_load((const v4f*)base);  // b128, TH=NT
        s += v;
        q += v * v;
        base += D_;
    }
    const size_t o = ((size_t)(seg * B_ + b)) * D_ + 4 * j;
    *(v4f*)(psum + o) = s;
    *(v4f*)(psq  + o) = q;
}

// ---------------------------------------------------------------------------
// Pass 3: reduce the 16 segment partials with V_WMMA_F32_16X16X4_F32.
// A = all-ones 16x4  =>  D[m,n] = C[m,n] + sum_k B[k,n]; four chained WMMAs
// sum all 16 segments in f32 (every output row holds the same totals, so any
// K-row -> lane permutation of B is equivalent). One full wave32 per
// (batch, 16-d chunk); EXEC is all ones as WMMA requires.
// Lanes 0-15 write the mean, lanes 16-31 the stddev half of the output.
// ---------------------------------------------------------------------------
__global__ void SP_finalize_kernel(const float* __restrict__ psum,
                                   const float* __restrict__ psq,
                                   const int* __restrict__ lens,
                                   float* __restrict__ out) {
    const int blk    = blockIdx.x;                 // 0 .. B*(D/16)-1
    const int b      = blk / (D_ / 16);
    const int dchunk = blk % (D_ / 16);
    const int lane   = threadIdx.x;                // 0..31
    const int half   = lane >> 4;
    const int d      = dchunk * 16 + (lane & 15);

    const v2f a = {1.0f, 1.0f};                    // all-ones A matrix
    v8f csum = {};
    v8f csq  = {};
#pragma unroll
    for (int s0 = 0; s0 < S_; s0 += 4) {
        const int k0 = s0 + 2 * half;              // this lane-half carries rows k0, k0+1
        const size_t i0 = ((size_t)((k0)     * B_ + b)) * D_ + d;
        const size_t i1 = ((size_t)((k0 + 1) * B_ + b)) * D_ + d;
        const v2f bs = { psum[i0], psum[i1] };
        const v2f bq = { psq[i0],  psq[i1]  };
        csum = __builtin_amdgcn_wmma_f32_16x16x4_f32(
            false, a, false, bs, (short)0, csum, false, false);
        csq  = __builtin_amdgcn_wmma_f32_16x16x4_f32(
            false, a, false, bq, (short)0, csq, false, false);
    }
    // Row 0 (lanes 0-15) and row 8 (lanes 16-31) both hold the full K-sum.
    const float sum  = csum[0];
    const float sq   = csq[0];
    const float inv  = 1.0f / (float)lens[b];
    const float mean = sum * inv;
    const float sd   = sqrtf(sq * inv - mean * mean + 1e-6f);
    out[(size_t)b * (2 * D_) + half * D_ + d] = half ? sd : mean;
}

// ---------------------------------------------------------------------------
extern "C" void kernel_launch(void* const* d_in, const int* in_sizes, int n_in,
                              void* d_out, int out_size, void* d_ws, size_t ws_size,
                              hipStream_t stream) {
    const float* feat = (const float*)d_in[0];   // [B, T, D] f32
    const float* mask = (const float*)d_in[1];   // [B, T, 1] f32
    float* out  = (float*)d_out;                 // [B, 2D] f32
    float* wsf  = (float*)d_ws;
    float* psum = wsf;
    float* psq  = wsf + PSQ_OFS;
    int*   lens = (int*)(wsf + LEN_OFS);

    SP_lengths_kernel<<<B_, 256, 0, stream>>>(mask, lens);
    SP_partials_kernel<<<dim3(S_, B_), 192, 0, stream>>>(feat, lens, psum, psq);
    SP_finalize_kernel<<<B_ * (D_ / 16), 32, 0, stream>>>(psum, psq, lens, out);
}